// WaveletDWTLayer_81956565943140
// MI455X (gfx1250) — compile-verified
//
#include <hip/hip_runtime.h>
#include <stdint.h>

// ---------------- problem constants ----------------
#define L        4096      // signal length
#define NC       2051      // cA/cD length = (L+8-1)//2
#define LOUT3    8186      // R length = 2*L - 6
#define NTH      256       // threads per block (8 wave32)
#define FILLER_F 10.1f

// db4 reconstruction filters (REC_LO given; REC_HI[k] = (-1)^k * REC_LO[7-k])
// DEC_LO[k] = REC_LO[7-k], DEC_HI[k] = REC_HI[7-k]  (indexed reversed below)
static constexpr float RL[8] = {
     0.23037781330885523f,  0.7148465705525415f,   0.6308807679295904f,  -0.02798376941698385f,
    -0.18703481171888114f,  0.030841381835986965f, 0.032883011666982945f, -0.010597401784997278f };
static constexpr float RH[8] = {
    -0.010597401784997278f, -0.032883011666982945f, 0.030841381835986965f, 0.18703481171888114f,
    -0.02798376941698385f,  -0.6308807679295904f,   0.7148465705525415f,  -0.23037781330885523f };

typedef unsigned int u32x4 __attribute__((ext_vector_type(4)));
typedef int          i32x8 __attribute__((ext_vector_type(8)));
typedef int          i32x4 __attribute__((ext_vector_type(4)));
typedef float        f32x4 __attribute__((ext_vector_type(4)));
typedef float        f32x2 __attribute__((ext_vector_type(2)));

// ---- Tensor Data Mover: DMA one contiguous row of `nelem` f32 from global to LDS ----
// D# per CDNA5 ISA: group0 = {count=1 | lds_addr | global_addr | type=2},
// group1 = {mask=0,data_size=4B | tensor_dim0=nelem, tensor_dim1=1, tile_dim0=nelem, tile_dim1=1,
//           tensor_dim0_stride=nelem}. Groups 2/3 unused (<=2D tensor).
__device__ __forceinline__ void tdm_load_row_f32(const float* src, void* lds_dst, int nelem) {
    uint64_t ga     = (uint64_t)(uintptr_t)src;
    uint32_t ldsoff = (uint32_t)(uintptr_t)lds_dst;   // low 32 bits of generic ptr = LDS byte offset
    u32x4 g0;
    g0[0] = 1u;                                               // count=1, is_restore=0, gather off
    g0[1] = ldsoff;                                           // lds_addr
    g0[2] = (uint32_t)ga;                                     // global_addr[31:0]
    g0[3] = ((uint32_t)(ga >> 32) & 0x01FFFFFFu) | (2u << 30);// global_addr[56:32] | type=2
    i32x8 g1;
    g1[0] = (int)(2u << 16);                                  // workgroup_mask=0, data_size=2 (4B)
    g1[1] = (int)(((uint32_t)nelem & 0xFFFFu) << 16);         // tensor_dim0[15:0] << 16
    g1[2] = (int)(((uint32_t)nelem >> 16) | (1u << 16));      // tensor_dim0[31:16] | tensor_dim1[15:0]=1
    g1[3] = (int)((uint32_t)nelem << 16);                     // tensor_dim1[31:16]=0 | tile_dim0
    g1[4] = 1;                                                // tile_dim1=1, tile_dim2=0
    g1[5] = nelem;                                            // tensor_dim0_stride[31:0]
    g1[6] = 0;
    g1[7] = 0;
    i32x4 z4 = {0, 0, 0, 0};
    i32x8 z8 = {0, 0, 0, 0, 0, 0, 0, 0};
    // clang-23 / therock-10.0 6-arg form: (g0, g1, g2, g3, g4, cpol)
    __builtin_amdgcn_tensor_load_to_lds(g0, g1, z4, z4, z8, 0);
}

__global__ __launch_bounds__(NTH)
void wavelet_fused_kernel(const float* __restrict__ x1, const float* __restrict__ x2,
                          float* __restrict__ out1, float* __restrict__ out2,
                          float* __restrict__ out3)
{
    __shared__ __align__(16) float s_x1[L];       // input signal row
    __shared__ __align__(16) float s_y [L];       // x2 row
    __shared__ __align__(16) float s_cD[NC + 1];  // detail coeffs (stay in LDS)
    __shared__ __align__(16) float s_a [L];       // a = idwt(0, cD) intermediate (never hits HBM)

    const int b   = blockIdx.x;
    const int tid = threadIdx.x;
    const float* x1row = x1 + (size_t)b * L;
    const float* yrow  = x2 + (size_t)b * L;
    float* o1 = out1 + (size_t)b * NC;        // cA row
    float* o2 = out2 + (size_t)b * (2 * L);   // cDs row: [y(4096) ; cD padded(4096)]
    float* o3 = out3 + (size_t)b * LOUT3;     // R row

    // ---- Stage 0: TDM async DMA of both input rows into LDS (issued once, wave 0) ----
    if (tid == 0) {
        tdm_load_row_f32(x1row, (void*)s_x1, L);
        tdm_load_row_f32(yrow,  (void*)s_y,  L);
    }
    __builtin_amdgcn_s_wait_tensorcnt(0);   // no-op for waves with TENSORcnt==0
    __syncthreads();

    // ---- Stage 1: DWT.  ext[i] = x[mirror(i-6)] (symmetric pad 7/7, drop first) ----
    // cA[j] = sum_k RL[k]*ext[2j+k];  cD[j] = sum_k RH[k]*ext[2j+k]
    for (int j = tid; j < NC; j += NTH) {
        const int base = 2 * j - 6;
        float lo = 0.0f, hi = 0.0f;
        #pragma unroll
        for (int k = 0; k < 8; ++k) {
            int m = base + k;
            m = (m < 0) ? (-1 - m) : ((m >= L) ? (2 * L - 1 - m) : m);
            const float v = s_x1[m];
            lo = fmaf(v, RL[k], lo);
            hi = fmaf(v, RH[k], hi);
        }
        __builtin_nontemporal_store(lo, &o1[j]);   // cA: streaming write-once
        s_cD[j] = hi;
    }
    __syncthreads();

    // ---- Stage 2a: cDs output = [y ; cD padded with FILLER], 128-bit NT stores ----
    {
        const f32x4* yv = (const f32x4*)s_y;
        for (int v = tid; v < (2 * L) / 4; v += NTH) {
            const int j = 4 * v;
            f32x4 val;
            if (j < L) {
                val = yv[v];
            } else {
                const int jj = j - L;
                val[0] = (jj + 0 < NC) ? s_cD[jj + 0] : FILLER_F;
                val[1] = (jj + 1 < NC) ? s_cD[jj + 1] : FILLER_F;
                val[2] = (jj + 2 < NC) ? s_cD[jj + 2] : FILLER_F;
                val[3] = (jj + 3 < NC) ? s_cD[jj + 3] : FILLER_F;
            }
            __builtin_nontemporal_store(val, (f32x4*)o2 + v);
        }
    }

    // ---- Stage 2b: a = idwt(0, cD)  (upsample-by-2 synthesis; indices provably in-range)
    // even i: sum_t cD[i/2+t]     * DEC_HI[2t+1] = RH[6],RH[4],RH[2],RH[0]
    // odd  i: sum_t cD[(i-1)/2+t] * DEC_HI[2t]   = RH[7],RH[5],RH[3],RH[1]
    for (int i = tid; i < L; i += NTH) {
        const int h = i >> 1;
        const float c0 = s_cD[h], c1 = s_cD[h + 1], c2 = s_cD[h + 2], c3 = s_cD[h + 3];
        float acc;
        if (i & 1) {
            acc =      c0 * RH[7];
            acc = fmaf(c1,  RH[5], acc);
            acc = fmaf(c2,  RH[3], acc);
            acc = fmaf(c3,  RH[1], acc);
        } else {
            acc =      c0 * RH[6];
            acc = fmaf(c1,  RH[4], acc);
            acc = fmaf(c2,  RH[2], acc);
            acc = fmaf(c3,  RH[0], acc);
        }
        s_a[i] = acc;
    }
    __syncthreads();

    // ---- Stage 3: R = idwt(a, y), len 8186 = 4093 x float2 (rows are 8B aligned) ----
    // pair v -> elements (2v even, 2v+1 odd), window index base = v, v+3 <= 4095 always.
    // even: sum_t a[v+t]*DEC_LO[2t+1] + y[v+t]*DEC_HI[2t+1]  (DEC_LO[2t+1]=RL[6],RL[4],RL[2],RL[0])
    // odd : sum_t a[v+t]*DEC_LO[2t]   + y[v+t]*DEC_HI[2t]    (DEC_LO[2t]  =RL[7],RL[5],RL[3],RL[1])
    {
        f32x2* o3v = (f32x2*)o3;
        for (int v = tid; v < LOUT3 / 2; v += NTH) {
            const float a0 = s_a[v], a1 = s_a[v + 1], a2 = s_a[v + 2], a3 = s_a[v + 3];
            const float y0 = s_y[v], y1 = s_y[v + 1], y2 = s_y[v + 2], y3 = s_y[v + 3];
            f32x2 r;
            float e =      a0 * RL[6];
            e = fmaf(a1, RL[4], e);  e = fmaf(a2, RL[2], e);  e = fmaf(a3, RL[0], e);
            e = fmaf(y0, RH[6], e);  e = fmaf(y1, RH[4], e);
            e = fmaf(y2, RH[2], e);  e = fmaf(y3, RH[0], e);
            r[0] = e;
            float o =      a0 * RL[7];
            o = fmaf(a1, RL[5], o);  o = fmaf(a2, RL[3], o);  o = fmaf(a3, RL[1], o);
            o = fmaf(y0, RH[7], o);  o = fmaf(y1, RH[5], o);
            o = fmaf(y2, RH[3], o);  o = fmaf(y3, RH[1], o);
            r[1] = o;
            __builtin_nontemporal_store(r, &o3v[v]);
        }
    }
}

extern "C" void kernel_launch(void* const* d_in, const int* in_sizes, int n_in,
                              void* d_out, int out_size, void* d_ws, size_t ws_size,
                              hipStream_t stream) {
    (void)n_in; (void)out_size; (void)d_ws; (void)ws_size;
    const float* x1 = (const float*)d_in[0];   // [B,1,L]
    const float* x2 = (const float*)d_in[1];   // [B,1,1,L]
    // d_in[2] (x3) is unused by the reference.
    const int B = in_sizes[0] / L;             // 4096
    float* out = (float*)d_out;
    float* o1 = out;                                   // cA   [B,1,NC]
    float* o2 = o1 + (size_t)B * NC;                   // cDs  [B,2,1,L]
    float* o3 = o2 + (size_t)B * 2 * L;                // R    [B,1,1,LOUT3]
    hipLaunchKernelGGL(wavelet_fused_kernel, dim3(B), dim3(NTH), 0, stream,
                       x1, x2, o1, o2, o3);
}